// FusionAndClassifier_41755672051947
// MI455X (gfx1250) — compile-verified
//
#include <hip/hip_runtime.h>
#include <math.h>

// ---------------- problem constants ----------------
#define NN 262144
#define BB 512
#define HH 128
#define DD 256
#define CC 10

typedef __attribute__((ext_vector_type(2))) float v2f;
typedef __attribute__((ext_vector_type(8))) float v8f;

// XOR-swizzled LDS addressing: row stride 256 floats, bank-conflict free for
// float4 staging, A-fragment loads and C-layout scatter.
__device__ __forceinline__ int swz(int r, int c)
{
    return (r << 8) + (c ^ ((r << 2) & 60));
}

// ---------------------------------------------------------------------------
// GEMM core: A = 32x256 fp32 tile in swizzled LDS, B = 256x256 weights packed
// as b128 fragment-pairs, acc = 2 row-tiles x 16 col-tiles of 16x16 fp32.
// Each B fragment (loaded once) feeds two WMMAs (row-tile 0 and 1).
// Fragment layouts per CDNA5 ISA 7.12.2:
//   A 16x4:  lanes 0-15: M=lane, K=4k+{0,1}; lanes 16-31: K=4k+{2,3}
//   B 4x16:  lanes 0-15: N=lane, K=4k+{0,1}; lanes 16-31: K=4k+{2,3}
//   C/D:     vgpr i -> M = i (+8 for lanes 16-31), N = lane&15
// ---------------------------------------------------------------------------
__device__ __forceinline__ void gemm32x256(const float* __restrict__ lds_a,
                                           const float* __restrict__ packedW,
                                           v8f (&acc)[32], int lane)
{
    const int lo = lane & 15;
    const int hi = lane >> 4;
    const float4* __restrict__ bp = (const float4*)packedW;
    for (int k = 0; k < 64; ++k) {                       // K chunks of 4
        v2f a0 = *(const v2f*)(lds_a + swz(lo,      4 * k + 2 * hi));
        v2f a1 = *(const v2f*)(lds_a + swz(lo + 16, 4 * k + 2 * hi));
        const float4* bk = bp + k * 256 + lane;
#pragma unroll
        for (int p = 0; p < 8; ++p) {
            float4 q = bk[p * 32];                       // one b128 = 2 B frags
            v2f b0; b0.x = q.x; b0.y = q.y;
            v2f b1; b1.x = q.z; b1.y = q.w;
            acc[2 * p] = __builtin_amdgcn_wmma_f32_16x16x4_f32(
                false, a0, false, b0, (short)0, acc[2 * p], false, false);
            acc[2 * p + 1] = __builtin_amdgcn_wmma_f32_16x16x4_f32(
                false, a0, false, b1, (short)0, acc[2 * p + 1], false, false);
            acc[16 + 2 * p] = __builtin_amdgcn_wmma_f32_16x16x4_f32(
                false, a1, false, b0, (short)0, acc[16 + 2 * p], false, false);
            acc[16 + 2 * p + 1] = __builtin_amdgcn_wmma_f32_16x16x4_f32(
                false, a1, false, b1, (short)0, acc[16 + 2 * p + 1], false, false);
        }
    }
}

// ---------------------------------------------------------------------------
// Kernel 0: repack gate_W1, gate_W2, pool_W into b128 fragment-pair order.
// float4 at ((k*8+p)*32+lane) = { W[kk][32p+lo], W[kk+1][32p+lo],
//                                 W[kk][32p+16+lo], W[kk+1][32p+16+lo] }
// with kk = 4k + ((lane<16)?0:2), lo = lane&15.
// ---------------------------------------------------------------------------
__global__ void repack_kernel(const float* __restrict__ W1,
                              const float* __restrict__ W2,
                              const float* __restrict__ W3,
                              float* __restrict__ pk)
{
    int t = blockIdx.x * 256 + threadIdx.x;
    if (t >= 3 * 16384) return;
    int m    = t >> 14;
    int rem  = t & 16383;
    int k    = rem >> 8;
    int rem2 = rem & 255;
    int p    = rem2 >> 5;
    int lane = rem2 & 31;
    const float* W = (m == 0) ? W1 : ((m == 1) ? W2 : W3);
    int roff = (lane < 16) ? 0 : 2;
    int lo   = lane & 15;
    int c0   = 32 * p + lo;
    int c1   = c0 + 16;
    int kk   = 4 * k + roff;
    float4 v;
    v.x = W[kk * DD + c0];
    v.y = W[(kk + 1) * DD + c0];
    v.z = W[kk * DD + c1];
    v.w = W[(kk + 1) * DD + c1];
    ((float4*)(pk + (size_t)m * 65536))[(k * 8 + p) * 32 + lane] = v;
}

// ---------------------------------------------------------------------------
// Kernel 1: per-node fused pipeline, one wave per 32-node tile.
// ---------------------------------------------------------------------------
__global__ __launch_bounds__(32) void node_kernel(
    const float* __restrict__ Hi, const float* __restrict__ He,
    const float* __restrict__ pk,
    const float* __restrict__ gb1, const float* __restrict__ gb2,
    const float* __restrict__ pb,  const float* __restrict__ pvec,
    float* __restrict__ Hf_out, float* __restrict__ s_out)
{
    __shared__ float zbuf[32 * 256];   // Z tile, later H_fused (swizzled)
    __shared__ float hbuf[32 * 256];   // gelu hidden tile (swizzled)

    const int lane  = threadIdx.x;
    const int lo    = lane & 15;
    const int hi    = lane >> 4;
    const int node0 = blockIdx.x * 32;

    // ---- stage Z tile: cols [0,128) = H_intra, [128,256) = H_inter ----
#pragma unroll 4
    for (int r = 0; r < 32; ++r) {
        float4 a = ((const float4*)(Hi + (size_t)(node0 + r) * HH))[lane];
        float4 b = ((const float4*)(He + (size_t)(node0 + r) * HH))[lane];
        int sw = lane ^ (r & 15);                  // float4-unit swizzle
        ((float4*)zbuf)[r * 64 + sw]      = a;     // cols  4*lane..
        ((float4*)zbuf)[r * 64 + 32 + sw] = b;     // cols 128+4*lane..
    }
    __syncthreads();

    v8f acc[32];
    const v8f vzero = {};

    // ---------------- GEMM1: h = gelu(Z@W1 + b1) ----------------
#pragma unroll
    for (int i = 0; i < 32; ++i) acc[i] = vzero;
    gemm32x256(zbuf, pk, acc, lane);
#pragma unroll
    for (int t = 0; t < 2; ++t) {
#pragma unroll
        for (int nt = 0; nt < 16; ++nt) {
            float bias = gb1[nt * 16 + lo];
#pragma unroll
            for (int i = 0; i < 8; ++i) {
                float x = acc[t * 16 + nt][i] + bias;
                float g = 0.5f * x * (1.0f + erff(x * 0.70710678118654752f));
                hbuf[swz(16 * t + 8 * hi + i, nt * 16 + lo)] = g;
            }
        }
    }
    __syncthreads();

    // ---------------- GEMM2: g = sigmoid(h@W2 + b2); Hf = g*Z ----------------
#pragma unroll
    for (int i = 0; i < 32; ++i) acc[i] = vzero;
    gemm32x256(hbuf, pk + 65536, acc, lane);
#pragma unroll
    for (int t = 0; t < 2; ++t) {
#pragma unroll
        for (int nt = 0; nt < 16; ++nt) {
            float bias = gb2[nt * 16 + lo];
#pragma unroll
            for (int i = 0; i < 8; ++i) {
                float x = acc[t * 16 + nt][i] + bias;
                float g = 1.0f / (1.0f + expf(-x));
                int r = 16 * t + 8 * hi + i;
                int c = nt * 16 + lo;
                float z = zbuf[swz(r, c)];
                float f = g * z;
                zbuf[swz(r, c)] = f;                               // Hf -> LDS
                Hf_out[(size_t)(node0 + r) * DD + c] = f;          // Hf -> out
            }
        }
    }
    __syncthreads();

    // ---------------- GEMM3: s = tanh(Hf@poolW + pb) . pvec ----------------
#pragma unroll
    for (int i = 0; i < 32; ++i) acc[i] = vzero;
    gemm32x256(zbuf, pk + 2 * 65536, acc, lane);

    float rs[2][8];
#pragma unroll
    for (int t = 0; t < 2; ++t)
#pragma unroll
        for (int i = 0; i < 8; ++i) rs[t][i] = 0.0f;
#pragma unroll
    for (int t = 0; t < 2; ++t) {
#pragma unroll
        for (int nt = 0; nt < 16; ++nt) {
            float bias = pb[nt * 16 + lo];
            float w    = pvec[nt * 16 + lo];
#pragma unroll
            for (int i = 0; i < 8; ++i)
                rs[t][i] += tanhf(acc[t * 16 + nt][i] + bias) * w;
        }
    }
    // reduce over the 16 lanes of each half (cols) -> per-row scalar
#pragma unroll
    for (int off = 1; off < 16; off <<= 1) {
#pragma unroll
        for (int t = 0; t < 2; ++t)
#pragma unroll
            for (int i = 0; i < 8; ++i)
                rs[t][i] += __shfl_xor(rs[t][i], off, 32);
    }
    if (lo == 0) {
#pragma unroll
        for (int t = 0; t < 2; ++t)
#pragma unroll
            for (int i = 0; i < 8; ++i)
                s_out[node0 + 16 * t + 8 * hi + i] = rs[t][i];
    }
}

// ---------------------------------------------------------------------------
// Kernel 2: per-graph segment softmax + attention pooling + classifier.
// ---------------------------------------------------------------------------
__device__ __forceinline__ int lower_bound_dev(const int* __restrict__ arr,
                                               int n, int key)
{
    int loi = 0, hii = n;
    while (loi < hii) {
        int mid = (loi + hii) >> 1;
        if (arr[mid] < key) loi = mid + 1; else hii = mid;
    }
    return loi;
}

__global__ __launch_bounds__(256) void seg_kernel(
    const float* __restrict__ s_arr, const float* __restrict__ Hf,
    const int* __restrict__ batch,
    const float* __restrict__ cW1, const float* __restrict__ cb1,
    const float* __restrict__ cW2, const float* __restrict__ cb2,
    float* __restrict__ logits, float* __restrict__ emb_out,
    float* __restrict__ attn_out)
{
    __shared__ float red[256];
    __shared__ float lds_a[256];
    __shared__ float lds_emb[256];
    __shared__ float lds_t1[256];

    const int b   = blockIdx.x;
    const int tid = threadIdx.x;
    const int start = lower_bound_dev(batch, NN, b);
    const int end   = lower_bound_dev(batch, NN, b + 1);

    // ---- segment max ----
    float mx = -INFINITY;
    for (int j = start + tid; j < end; j += 256) mx = fmaxf(mx, s_arr[j]);
    red[tid] = mx; __syncthreads();
    for (int o = 128; o > 0; o >>= 1) {
        if (tid < o) red[tid] = fmaxf(red[tid], red[tid + o]);
        __syncthreads();
    }
    mx = red[0]; __syncthreads();

    // ---- segment sum of exp ----
    float sm = 0.0f;
    for (int j = start + tid; j < end; j += 256) sm += expf(s_arr[j] - mx);
    red[tid] = sm; __syncthreads();
    for (int o = 128; o > 0; o >>= 1) {
        if (tid < o) red[tid] += red[tid + o];
        __syncthreads();
    }
    const float denom = red[0] + 1e-12f; __syncthreads();

    // ---- attn + weighted segment-sum (thread tid owns column tid) ----
    float emb = 0.0f;
    for (int cs = start; cs < end; cs += 256) {
        int j = cs + tid;
        float a = 0.0f;
        if (j < end) {
            a = expf(s_arr[j] - mx) / denom;
            attn_out[j] = a;
        }
        lds_a[tid] = a; __syncthreads();
        int cnt = min(256, end - cs);
        for (int t = 0; t < cnt; ++t)
            emb += lds_a[t] * Hf[(size_t)(cs + t) * DD + tid];
        __syncthreads();
    }
    emb_out[(size_t)b * DD + tid] = emb;
    lds_emb[tid] = emb; __syncthreads();

    // ---- classifier: relu(emb@W1+b1)@W2 + b2 ----
    float t1 = cb1[tid];
    for (int k = 0; k < DD; ++k)
        t1 += lds_emb[k] * cW1[(size_t)k * DD + tid];
    t1 = fmaxf(t1, 0.0f);
    lds_t1[tid] = t1; __syncthreads();

    if (tid < CC) {
        float o = cb2[tid];
        for (int k = 0; k < DD; ++k)
            o += lds_t1[k] * cW2[(size_t)k * CC + tid];
        logits[(size_t)b * CC + tid] = o;
    }
}

// ---------------------------------------------------------------------------
extern "C" void kernel_launch(void* const* d_in, const int* in_sizes, int n_in,
                              void* d_out, int out_size, void* d_ws, size_t ws_size,
                              hipStream_t stream)
{
    const float* Hi   = (const float*)d_in[0];
    const float* He   = (const float*)d_in[1];
    const int*   batch= (const int*)  d_in[2];
    const float* gW1  = (const float*)d_in[3];
    const float* gb1  = (const float*)d_in[4];
    const float* gW2  = (const float*)d_in[5];
    const float* gb2  = (const float*)d_in[6];
    const float* pW   = (const float*)d_in[7];
    const float* pb   = (const float*)d_in[8];
    const float* pv   = (const float*)d_in[9];
    const float* cW1  = (const float*)d_in[10];
    const float* cb1  = (const float*)d_in[11];
    const float* cW2  = (const float*)d_in[12];
    const float* cb2  = (const float*)d_in[13];

    float* out    = (float*)d_out;
    float* logits = out;                                   // [512*10]
    float* emb    = out + BB * CC;                         // [512*256]
    float* attn   = emb + BB * DD;                         // [N]
    float* Hf     = attn + NN;                             // [N*256]

    float* ws    = (float*)d_ws;
    float* pk    = ws;                                     // 3*65536 floats
    float* s_arr = ws + 3 * 65536;                         // N floats

    repack_kernel<<<(3 * 16384 + 255) / 256, 256, 0, stream>>>(gW1, gW2, pW, pk);
    node_kernel<<<NN / 32, 32, 0, stream>>>(Hi, He, pk, gb1, gb2, pb, pv, Hf, s_arr);
    seg_kernel<<<BB, 256, 0, stream>>>(s_arr, Hf, batch, cW1, cb1, cW2, cb2,
                                       logits, emb, attn);
}